// AttractorPooling_41824391528936
// MI455X (gfx1250) — compile-verified
//
#include <hip/hip_runtime.h>
#include <math.h>

typedef float v2f __attribute__((ext_vector_type(2)));
typedef float v8f __attribute__((ext_vector_type(8)));

#define N_TRAJ 4096
#define BATCH  8
#define NRAD   20
#define EPSF   1e-8f

// One workgroup = one (batch, 16-row i-tile). 8 waves split the 256 j-tiles.
// Whole per-batch trajectory lives in LDS (48 KB << 320 KB/WGP).
__global__ __launch_bounds__(256) void corrdim_pairs_kernel(
    const float* __restrict__ traj,     // [B, N, 3]
    const float* __restrict__ radii,    // [NRAD]
    unsigned int* __restrict__ counts)  // [B, NRAD] accumulators (pre-zeroed)
{
    __shared__ float sx[N_TRAJ];
    __shared__ float sy[N_TRAJ];
    __shared__ float sz[N_TRAJ];
    __shared__ unsigned int wgcnt[NRAD];

    const int tid   = threadIdx.x;
    const int b     = blockIdx.y;
    const int itile = blockIdx.x;        // 0..255
    const int i0    = itile * 16;

    if (tid < NRAD) wgcnt[tid] = 0u;

    // Stage this batch's trajectory into LDS.
    const float* tb = traj + (size_t)b * N_TRAJ * 3;
    __builtin_prefetch(tb, 0, 0);        // global_prefetch_b8
    for (int idx = tid; idx < N_TRAJ; idx += 256) {
        const float* p = tb + idx * 3;
        sx[idx] = p[0];
        sy[idx] = p[1];
        sz[idx] = p[2];
    }

    // radii^2 (uniform -> SGPRs). d < r  <=>  d^2 < r^2; EPS clamp can't flip it.
    float r2[NRAD];
#pragma unroll
    for (int k = 0; k < NRAD; ++k) { float r = radii[k]; r2[k] = r * r; }

    __syncthreads();

    const int lane = tid & 31;           // wave32
    const int wave = tid >> 5;           // 0..7
    const int half = lane >> 4;          // 0: K={0,1}; 1: K={2,3}
    const int lrow = lane & 15;

    // A operand: 16x4 f32 tile of rows i0..i0+15 (K = x,y,z,0), fixed per WG.
    // ISA layout: lanes 0-15 -> (K0,K1), lanes 16-31 -> (K2,K3) for row M=lane%16.
    v2f a;
    if (half == 0) { a.x = sx[i0 + lrow]; a.y = sy[i0 + lrow]; }
    else           { a.x = sz[i0 + lrow]; a.y = 0.0f;          }

    // Squared norms of this lane's 8 output rows (C/D layout: M = v + 8*half).
    float sqA[8];
#pragma unroll
    for (int v = 0; v < 8; ++v) {
        int gi = i0 + v + 8 * half;
        float x = sx[gi], y = sy[gi], z = sz[gi];
        sqA[v] = x * x + y * y + z * z;
    }

    // Per-WAVE scalar counters: each lane computes the same value via
    // popcount(ballot(...)), so these live in SGPRs and the per-radius
    // accumulate is v_cmp (VALU) + s_bcnt/s_add (SALU, co-issues).
    unsigned int cnt[NRAD];
#pragma unroll
    for (int k = 0; k < NRAD; ++k) cnt[k] = 0u;

    // Each wave handles j-tiles wave, wave+8, ... (uniform per wave -> EXEC
    // is all ones at every WMMA and every ballot, as required).
    for (int jt = wave; jt < N_TRAJ / 16; jt += 8) {
        const int j0 = jt * 16;

        // B operand: 4x16 tile of columns j0..j0+15 (same symmetric layout).
        v2f bv;
        if (half == 0) { bv.x = sx[j0 + lrow]; bv.y = sy[j0 + lrow]; }
        else           { bv.x = sz[j0 + lrow]; bv.y = 0.0f;          }

        v8f c = {0.f, 0.f, 0.f, 0.f, 0.f, 0.f, 0.f, 0.f};
        // D = A(16x4) x B(4x16) + 0 : Gram tile of dot products.
        c = __builtin_amdgcn_wmma_f32_16x16x4_f32(
                /*neg_a=*/false, a, /*neg_b=*/false, bv,
                /*c_mod=*/(short)0, c, /*reuse_a=*/false, /*reuse_b=*/false);

        const float xj = sx[j0 + lrow], yj = sy[j0 + lrow], zj = sz[j0 + lrow];
        const float sqB = xj * xj + yj * yj + zj * zj;
        const int gj = j0 + lrow;        // this lane's column index

#pragma unroll
        for (int v = 0; v < 8; ++v) {
            const int gi = i0 + v + 8 * half;
            float d2 = sqA[v] + sqB - 2.0f * c[v];
            if (gi == gj) d2 = 3.0e38f;  // exclude diagonal pairs
#pragma unroll
            for (int k = 0; k < NRAD; ++k) {
                unsigned int m = __builtin_amdgcn_ballot_w32(d2 < r2[k]);
                cnt[k] += (unsigned int)__builtin_popcount(m);
            }
        }
    }

    // Wave-uniform counters: one lane per wave commits to LDS, then block -> global.
    if (lane == 0) {
#pragma unroll
        for (int k = 0; k < NRAD; ++k)
            atomicAdd(&wgcnt[k], cnt[k]);
    }
    __syncthreads();
    if (tid < NRAD)
        atomicAdd(&counts[b * NRAD + tid], wgcnt[tid]);
}

// Tiny finalize: log C(r), mean finite-difference slope, clamp.
__global__ __launch_bounds__(32) void corrdim_finalize_kernel(
    const unsigned int* __restrict__ counts,
    const float* __restrict__ radii,
    float* __restrict__ out)
{
    const int b = threadIdx.x;
    if (b >= BATCH) return;
    const float total = (float)N_TRAJ * (float)(N_TRAJ - 1);

    float logC[NRAD], logr[NRAD];
#pragma unroll
    for (int k = 0; k < NRAD; ++k) {
        logC[k] = logf((float)counts[b * NRAD + k] / total + EPSF);
        logr[k] = logf(radii[k] + EPSF);
    }
    float s = 0.0f;
#pragma unroll
    for (int k = 0; k < NRAD - 1; ++k)
        s += (logC[k + 1] - logC[k]) / (logr[k + 1] - logr[k]);
    s /= (float)(NRAD - 1);
    s = fminf(fmaxf(s, 0.1f), 3.0f);
    out[b] = s;
}

extern "C" void kernel_launch(void* const* d_in, const int* in_sizes, int n_in,
                              void* d_out, int out_size, void* d_ws, size_t ws_size,
                              hipStream_t stream) {
    const float* traj  = (const float*)d_in[0];   // [8, 4096, 3] f32
    const float* radii = (const float*)d_in[1];   // [20] f32
    unsigned int* counts = (unsigned int*)d_ws;   // [8][20] u32

    hipMemsetAsync(counts, 0, BATCH * NRAD * sizeof(unsigned int), stream);

    dim3 grid(N_TRAJ / 16, BATCH);                // 256 i-tiles x 8 batches
    corrdim_pairs_kernel<<<grid, 256, 0, stream>>>(traj, radii, counts);
    corrdim_finalize_kernel<<<1, 32, 0, stream>>>(counts, radii, (float*)d_out);
}